// LRU2_13477607375572
// MI455X (gfx1250) — compile-verified
//
#include <hip/hip_runtime.h>
#include <hip/hip_bf16.h>

// Problem constants (from reference)
#define L_SEQ   32768
#define H_DIM   256
#define N_DIM   512
#define T_CHUNK 128
#define C_CHUNKS (L_SEQ / T_CHUNK)   // 256

typedef float v2f __attribute__((ext_vector_type(2)));
typedef float v8f __attribute__((ext_vector_type(8)));

// ---------------------------------------------------------------------------
// K0: per-channel Lambda, gamma, power table Lambda^(t+1), Lambda^T
// ---------------------------------------------------------------------------
__global__ void lru_prep(const float* __restrict__ nu_log,
                         const float* __restrict__ theta_log,
                         float* __restrict__ gamma,
                         float* __restrict__ lamRe, float* __restrict__ lamIm,
                         float* __restrict__ lamTRe, float* __restrict__ lamTIm,
                         float* __restrict__ powRe, float* __restrict__ powIm) {
  int n = threadIdx.x;                      // 512 threads, one block
  float nu  = __expf(nu_log[n]);
  float th  = __expf(theta_log[n]);
  float mod = __expf(-nu);                  // |Lambda|
  float lr = mod * __cosf(th);
  float li = mod * __sinf(th);
  lamRe[n] = lr;
  lamIm[n] = li;
  gamma[n] = sqrtf(fmaxf(0.0f, 1.0f - mod * mod));
  // pow[t] = Lambda^(t+1)
  float pr = lr, pi = li;
  powRe[n] = pr;
  powIm[n] = pi;
  for (int t = 1; t < T_CHUNK; ++t) {
    float nr = pr * lr - pi * li;
    float ni = pr * li + pi * lr;
    pr = nr; pi = ni;
    powRe[(size_t)t * N_DIM + n] = pr;
    powIm[(size_t)t * N_DIM + n] = pi;
  }
  lamTRe[n] = pr;                           // Lambda^T
  lamTIm[n] = pi;
}

// ---------------------------------------------------------------------------
// K0b: pre-materialize Bnorm = gamma[n] * B  and  CnIm = -C_im.
// All three arrays are N*H == H*N == 131072 elements.
// ---------------------------------------------------------------------------
__global__ void lru_prescale(const float* __restrict__ Bre,
                             const float* __restrict__ Bim,
                             const float* __restrict__ Cim,
                             const float* __restrict__ gamma,
                             float* __restrict__ BnRe,
                             float* __restrict__ BnIm,
                             float* __restrict__ CnIm) {
  int idx = blockIdx.x * blockDim.x + threadIdx.x;   // 0 .. N*H-1
  int n = idx / H_DIM;                               // B is [N,H]
  float g = gamma[n];
  BnRe[idx] = g * Bre[idx];
  BnIm[idx] = g * Bim[idx];
  CnIm[idx] = -Cim[idx];                             // C is [H,N], same count
}

// ---------------------------------------------------------------------------
// K1: Bu_re/Bu_im [L,N] = x [L,H] @ Bnorm^T  via WMMA f32 16x16x4.
// One wave computes a 64x16 output tile (4 row-tiles sharing B fragments).
// A 16x4 f32 layout: lane m = lane&15, k = 2*(lane>>4) + vgpr
// B 4x16 f32 layout: lane n = lane&15, k = 2*(lane>>4) + vgpr
// C 16x16 f32 layout: vgpr r -> M = r + 8*(lane>>4), N = lane&15
// ---------------------------------------------------------------------------
__global__ void lru_gemm1(const float* __restrict__ x,
                          const float* __restrict__ BnRe,
                          const float* __restrict__ BnIm,
                          float* __restrict__ BuRe,
                          float* __restrict__ BuIm) {
  const int lane = threadIdx.x & 31;
  const int wave = threadIdx.x >> 5;
  const int tile = blockIdx.x * 4 + wave;   // tiles = (L/64) * (N/16) = 512*32
  const int trg = tile >> 5;                // 64-row group (along L)
  const int tc  = tile & 31;                // col tile (along N)

  const int m  = lane & 15;
  const int kh = (lane >> 4) << 1;          // 0 or 2
  const int row0 = trg * 64 + m;            // A row, sub-tile 0
  const int ncol = tc * 16 + m;             // B column (global n)

  const float* a0 = x + (size_t)row0 * H_DIM + kh;
  const float* a1 = a0 + (size_t)16 * H_DIM;
  const float* a2 = a0 + (size_t)32 * H_DIM;
  const float* a3 = a0 + (size_t)48 * H_DIM;
  const float* brp = BnRe + (size_t)ncol * H_DIM + kh;
  const float* bip = BnIm + (size_t)ncol * H_DIM + kh;

  v8f aR0 = {}, aR1 = {}, aR2 = {}, aR3 = {};
  v8f aI0 = {}, aI1 = {}, aI2 = {}, aI3 = {};
  for (int kk = 0; kk < H_DIM; kk += 4) {
    v2f br = *(const v2f*)(brp + kk);
    v2f bi = *(const v2f*)(bip + kk);
    v2f f0 = *(const v2f*)(a0 + kk);
    v2f f1 = *(const v2f*)(a1 + kk);
    v2f f2 = *(const v2f*)(a2 + kk);
    v2f f3 = *(const v2f*)(a3 + kk);
    aR0 = __builtin_amdgcn_wmma_f32_16x16x4_f32(false, f0, false, br, (short)0, aR0, false, false);
    aI0 = __builtin_amdgcn_wmma_f32_16x16x4_f32(false, f0, false, bi, (short)0, aI0, false, false);
    aR1 = __builtin_amdgcn_wmma_f32_16x16x4_f32(false, f1, false, br, (short)0, aR1, false, false);
    aI1 = __builtin_amdgcn_wmma_f32_16x16x4_f32(false, f1, false, bi, (short)0, aI1, false, false);
    aR2 = __builtin_amdgcn_wmma_f32_16x16x4_f32(false, f2, false, br, (short)0, aR2, false, false);
    aI2 = __builtin_amdgcn_wmma_f32_16x16x4_f32(false, f2, false, bi, (short)0, aI2, false, false);
    aR3 = __builtin_amdgcn_wmma_f32_16x16x4_f32(false, f3, false, br, (short)0, aR3, false, false);
    aI3 = __builtin_amdgcn_wmma_f32_16x16x4_f32(false, f3, false, bi, (short)0, aI3, false, false);
  }

  const int col   = tc * 16 + (lane & 15);
  const int rbase = trg * 64 + ((lane >> 4) << 3);
#pragma unroll
  for (int r = 0; r < 8; ++r) {
    size_t i0 = (size_t)(rbase + r) * N_DIM + col;
    size_t i1 = (size_t)(rbase + 16 + r) * N_DIM + col;
    size_t i2 = (size_t)(rbase + 32 + r) * N_DIM + col;
    size_t i3 = (size_t)(rbase + 48 + r) * N_DIM + col;
    BuRe[i0] = aR0[r];  BuIm[i0] = aI0[r];
    BuRe[i1] = aR1[r];  BuIm[i1] = aI1[r];
    BuRe[i2] = aR2[r];  BuIm[i2] = aI2[r];
    BuRe[i3] = aR3[r];  BuIm[i3] = aI3[r];
  }
}

// ---------------------------------------------------------------------------
// K2: chunk-local inclusive scan s = Lambda*s + Bu, in place over Bu.
// grid = (N/256, C_CHUNKS), block = 256 (coalesced over n)
// ---------------------------------------------------------------------------
__global__ void lru_scan_local(const float* __restrict__ lamRe,
                               const float* __restrict__ lamIm,
                               float* __restrict__ BuRe,
                               float* __restrict__ BuIm,
                               float* __restrict__ clRe,
                               float* __restrict__ clIm) {
  const int n = blockIdx.x * blockDim.x + threadIdx.x;
  const int chunk = blockIdx.y;
  const float lr = lamRe[n];
  const float li = lamIm[n];
  size_t base = (size_t)chunk * T_CHUNK * N_DIM + n;
  float sr = 0.0f, si = 0.0f;
  for (int t = 0; t < T_CHUNK; ++t) {
    size_t idx = base + (size_t)t * N_DIM;
    float br = BuRe[idx];
    float bi = BuIm[idx];
    float nr = fmaf(lr, sr, fmaf(-li, si, br));
    float ni = fmaf(lr, si, fmaf( li, sr, bi));
    sr = nr; si = ni;
    BuRe[idx] = sr;
    BuIm[idx] = si;
  }
  clRe[(size_t)chunk * N_DIM + n] = sr;
  clIm[(size_t)chunk * N_DIM + n] = si;
}

// ---------------------------------------------------------------------------
// K3: exclusive prefix of chunk carries: carry' = Lambda^T * carry + local
// one block of 512 threads, sequential over 256 chunks
// ---------------------------------------------------------------------------
__global__ void lru_carry_prefix(const float* __restrict__ lamTRe,
                                 const float* __restrict__ lamTIm,
                                 const float* __restrict__ clRe,
                                 const float* __restrict__ clIm,
                                 float* __restrict__ ciRe,
                                 float* __restrict__ ciIm) {
  const int n = threadIdx.x;                // 512
  const float lr = lamTRe[n];
  const float li = lamTIm[n];
  float cr = 0.0f, ci = 0.0f;
  for (int c = 0; c < C_CHUNKS; ++c) {
    size_t idx = (size_t)c * N_DIM + n;
    ciRe[idx] = cr;                         // exclusive carry for chunk c
    ciIm[idx] = ci;
    float ar = clRe[idx];
    float ai = clIm[idx];
    float nr = fmaf(lr, cr, fmaf(-li, ci, ar));
    float ni = fmaf(lr, ci, fmaf( li, cr, ai));
    cr = nr; ci = ni;
  }
}

// ---------------------------------------------------------------------------
// K4: state[t] = local[t] + Lambda^(t+1) * carry_in(chunk)
// ---------------------------------------------------------------------------
__global__ void lru_apply_carry(const float* __restrict__ powRe,
                                const float* __restrict__ powIm,
                                const float* __restrict__ ciRe,
                                const float* __restrict__ ciIm,
                                float* __restrict__ SRe,
                                float* __restrict__ SIm) {
  const int n = blockIdx.x * blockDim.x + threadIdx.x;
  const int chunk = blockIdx.y;
  const float cr = ciRe[(size_t)chunk * N_DIM + n];
  const float ci = ciIm[(size_t)chunk * N_DIM + n];
  size_t base = (size_t)chunk * T_CHUNK * N_DIM + n;
  for (int t = 0; t < T_CHUNK; ++t) {
    size_t idx = base + (size_t)t * N_DIM;
    float pr = powRe[(size_t)t * N_DIM + n];
    float pi = powIm[(size_t)t * N_DIM + n];
    SRe[idx] = fmaf(pr, cr, fmaf(-pi, ci, SRe[idx]));
    SIm[idx] = fmaf(pr, ci, fmaf( pi, cr, SIm[idx]));
  }
}

// ---------------------------------------------------------------------------
// K5: y [L,H] = s_re @ C_re^T + s_im @ (-C_im)^T + D * x  via WMMA f32 16x16x4
// One wave computes a 64x16 output tile (4 row-tiles sharing C fragments).
// ---------------------------------------------------------------------------
__global__ void lru_gemm2(const float* __restrict__ SRe,
                          const float* __restrict__ SIm,
                          const float* __restrict__ Cre,
                          const float* __restrict__ CnIm,   // pre-negated C_im
                          const float* __restrict__ x,
                          const float* __restrict__ D,
                          float* __restrict__ y) {
  const int lane = threadIdx.x & 31;
  const int wave = threadIdx.x >> 5;
  const int tile = blockIdx.x * 4 + wave;   // tiles = (L/64) * (H/16) = 512*16
  const int trg = tile >> 4;                // 64-row group (along L)
  const int tc  = tile & 15;                // col tile (along H)

  const int m  = lane & 15;
  const int kh = (lane >> 4) << 1;
  const int row0 = trg * 64 + m;
  const int hcol = tc * 16 + m;

  const float* ar0 = SRe + (size_t)row0 * N_DIM + kh;
  const float* ar1 = ar0 + (size_t)16 * N_DIM;
  const float* ar2 = ar0 + (size_t)32 * N_DIM;
  const float* ar3 = ar0 + (size_t)48 * N_DIM;
  const float* ai0 = SIm + (size_t)row0 * N_DIM + kh;
  const float* ai1 = ai0 + (size_t)16 * N_DIM;
  const float* ai2 = ai0 + (size_t)32 * N_DIM;
  const float* ai3 = ai0 + (size_t)48 * N_DIM;
  const float* crp = Cre  + (size_t)hcol * N_DIM + kh;
  const float* cip = CnIm + (size_t)hcol * N_DIM + kh;

  v8f c0 = {}, c1 = {}, c2 = {}, c3 = {};
  for (int kk = 0; kk < N_DIM; kk += 4) {
    v2f br = *(const v2f*)(crp + kk);
    v2f bi = *(const v2f*)(cip + kk);
    v2f r0 = *(const v2f*)(ar0 + kk);
    v2f i0 = *(const v2f*)(ai0 + kk);
    v2f r1 = *(const v2f*)(ar1 + kk);
    v2f i1 = *(const v2f*)(ai1 + kk);
    v2f r2 = *(const v2f*)(ar2 + kk);
    v2f i2 = *(const v2f*)(ai2 + kk);
    v2f r3 = *(const v2f*)(ar3 + kk);
    v2f i3 = *(const v2f*)(ai3 + kk);
    c0 = __builtin_amdgcn_wmma_f32_16x16x4_f32(false, r0, false, br, (short)0, c0, false, false);
    c0 = __builtin_amdgcn_wmma_f32_16x16x4_f32(false, i0, false, bi, (short)0, c0, false, false);
    c1 = __builtin_amdgcn_wmma_f32_16x16x4_f32(false, r1, false, br, (short)0, c1, false, false);
    c1 = __builtin_amdgcn_wmma_f32_16x16x4_f32(false, i1, false, bi, (short)0, c1, false, false);
    c2 = __builtin_amdgcn_wmma_f32_16x16x4_f32(false, r2, false, br, (short)0, c2, false, false);
    c2 = __builtin_amdgcn_wmma_f32_16x16x4_f32(false, i2, false, bi, (short)0, c2, false, false);
    c3 = __builtin_amdgcn_wmma_f32_16x16x4_f32(false, r3, false, br, (short)0, c3, false, false);
    c3 = __builtin_amdgcn_wmma_f32_16x16x4_f32(false, i3, false, bi, (short)0, c3, false, false);
  }

  const int col   = tc * 16 + (lane & 15);
  const float d   = D[col];
  const int rbase = trg * 64 + ((lane >> 4) << 3);
#pragma unroll
  for (int r = 0; r < 8; ++r) {
    size_t i0 = (size_t)(rbase + r) * H_DIM + col;
    size_t i1 = (size_t)(rbase + 16 + r) * H_DIM + col;
    size_t i2 = (size_t)(rbase + 32 + r) * H_DIM + col;
    size_t i3 = (size_t)(rbase + 48 + r) * H_DIM + col;
    y[i0] = c0[r] + d * x[i0];
    y[i1] = c1[r] + d * x[i1];
    y[i2] = c2[r] + d * x[i2];
    y[i3] = c3[r] + d * x[i3];
  }
}

// ---------------------------------------------------------------------------
// Launcher
// ---------------------------------------------------------------------------
extern "C" void kernel_launch(void* const* d_in, const int* in_sizes, int n_in,
                              void* d_out, int out_size, void* d_ws, size_t ws_size,
                              hipStream_t stream) {
  (void)in_sizes; (void)n_in; (void)out_size; (void)ws_size;

  const float* x         = (const float*)d_in[0];   // [L,H]
  const float* nu_log    = (const float*)d_in[1];   // [N]
  const float* theta_log = (const float*)d_in[2];   // [N]
  const float* B_re      = (const float*)d_in[3];   // [N,H]
  const float* B_im      = (const float*)d_in[4];   // [N,H]
  const float* C_re      = (const float*)d_in[5];   // [H,N]
  const float* C_im      = (const float*)d_in[6];   // [H,N]
  const float* Dvec      = (const float*)d_in[7];   // [H]
  float* y               = (float*)d_out;           // [L,H]

  const size_t LN = (size_t)L_SEQ * N_DIM;
  const size_t NH = (size_t)N_DIM * H_DIM;
  float* ws    = (float*)d_ws;
  float* BuRe  = ws;                    ws += LN;                         // 64 MB
  float* BuIm  = ws;                    ws += LN;                         // 64 MB
  float* gamma = ws;                    ws += N_DIM;
  float* lamRe = ws;                    ws += N_DIM;
  float* lamIm = ws;                    ws += N_DIM;
  float* lamTRe = ws;                   ws += N_DIM;
  float* lamTIm = ws;                   ws += N_DIM;
  float* powRe = ws;                    ws += (size_t)T_CHUNK * N_DIM;
  float* powIm = ws;                    ws += (size_t)T_CHUNK * N_DIM;
  float* clRe  = ws;                    ws += (size_t)C_CHUNKS * N_DIM;
  float* clIm  = ws;                    ws += (size_t)C_CHUNKS * N_DIM;
  float* ciRe  = ws;                    ws += (size_t)C_CHUNKS * N_DIM;
  float* ciIm  = ws;                    ws += (size_t)C_CHUNKS * N_DIM;
  float* BnRe  = ws;                    ws += NH;                         // 512 KB
  float* BnIm  = ws;                    ws += NH;                         // 512 KB
  float* CnIm  = ws;                    ws += NH;                         // 512 KB

  // K0: Lambda / gamma / power tables
  lru_prep<<<1, N_DIM, 0, stream>>>(nu_log, theta_log, gamma,
                                    lamRe, lamIm, lamTRe, lamTIm, powRe, powIm);

  // K0b: Bnorm = gamma*B, CnIm = -C_im
  lru_prescale<<<NH / 256, 256, 0, stream>>>(B_re, B_im, C_im, gamma,
                                             BnRe, BnIm, CnIm);

  // K1: Bu = x @ Bnorm^T   (re & im)
  {
    int tiles = (L_SEQ / 64) * (N_DIM / 16);   // 16384 waves
    lru_gemm1<<<tiles / 4, 128, 0, stream>>>(x, BnRe, BnIm, BuRe, BuIm);
  }

  // K2: chunk-local scans (in place over Bu)
  lru_scan_local<<<dim3(N_DIM / 256, C_CHUNKS), 256, 0, stream>>>(
      lamRe, lamIm, BuRe, BuIm, clRe, clIm);

  // K3: carry prefix across chunks
  lru_carry_prefix<<<1, N_DIM, 0, stream>>>(lamTRe, lamTIm, clRe, clIm, ciRe, ciIm);

  // K4: add Lambda^(t+1) * carry to each chunk-local state
  lru_apply_carry<<<dim3(N_DIM / 256, C_CHUNKS), 256, 0, stream>>>(
      powRe, powIm, ciRe, ciIm, BuRe, BuIm);

  // K5: y = Re(states @ C^T) + D * x
  {
    int tiles = (L_SEQ / 64) * (H_DIM / 16);   // 8192 waves
    lru_gemm2<<<tiles / 4, 128, 0, stream>>>(BuRe, BuIm, C_re, CnIm, x, Dvec, y);
  }
}